// CRFLayer_63256278335780
// MI455X (gfx1250) — compile-verified
//
#include <hip/hip_runtime.h>

// ---------------------------------------------------------------------------
// CRF layer: B=128, S=512, NIN=512, T=64
//   emissions = x @ W + b          (WMMA f32 16x16x4 GEMM, 65536x64x512)
//   numerator[b] = sum_s em[b,s,y] + sum_s trans[y_{s-1}, y_s]
//   forward scan: lp[t] <- LSE_p(lp[p] + trans[p][t]) + em[s][t]
//   out = sum_b ( LSE_t(lp_final) - numerator[b] )
// ---------------------------------------------------------------------------

typedef __attribute__((ext_vector_type(2))) float v2f;
typedef __attribute__((ext_vector_type(8))) float v8f;

#define CRF_B   128
#define CRF_S   512
#define CRF_NIN 512
#define CRF_T   64
#define CRF_ROWS (CRF_B * CRF_S)   // 65536 GEMM rows

// -------------------- Kernel 1: emissions GEMM (WMMA) ----------------------
// One wave computes a 16-row x 64-col strip of emissions.
// 8 waves / block (256 thr), 4096 row tiles -> 512 blocks.
__global__ void crf_emissions_wmma(const float* __restrict__ x,   // (65536,512)
                                   const float* __restrict__ W,   // (512,64)
                                   const float* __restrict__ bias,// (64)
                                   float* __restrict__ em)        // (65536,64)
{
    const int lane    = threadIdx.x & 31;
    const int wave    = threadIdx.x >> 5;
    const int rowTile = blockIdx.x * 8 + wave;        // 0..4095
    const int hf      = lane >> 4;                    // 0 or 1
    const int lr      = lane & 15;

    const float* __restrict__ xRow = x + ((size_t)rowTile * 16 + lr) * CRF_NIN;

    v8f acc0 = {}, acc1 = {}, acc2 = {}, acc3 = {};

    for (int k = 0; k < CRF_NIN; k += 4) {
        const int ka = k + 2 * hf;                    // K pair owned by this half
        // A 16x4 tile: lane lr holds row, VGPR0/1 hold K=ka, ka+1 (contiguous)
        v2f a = *(const v2f*)(xRow + ka);

        // B 4x16 tiles: lane lr holds column, VGPR0/1 hold K=ka, ka+1
        const float* __restrict__ w0 = W + (size_t)ka * CRF_T;
        v2f b0, b1, b2, b3;
        b0.x = w0[lr];          b0.y = w0[CRF_T + lr];
        b1.x = w0[16 + lr];     b1.y = w0[CRF_T + 16 + lr];
        b2.x = w0[32 + lr];     b2.y = w0[CRF_T + 32 + lr];
        b3.x = w0[48 + lr];     b3.y = w0[CRF_T + 48 + lr];

        acc0 = __builtin_amdgcn_wmma_f32_16x16x4_f32(false, a, false, b0,
                                                     (short)0, acc0, false, false);
        acc1 = __builtin_amdgcn_wmma_f32_16x16x4_f32(false, a, false, b1,
                                                     (short)0, acc1, false, false);
        acc2 = __builtin_amdgcn_wmma_f32_16x16x4_f32(false, a, false, b2,
                                                     (short)0, acc2, false, false);
        acc3 = __builtin_amdgcn_wmma_f32_16x16x4_f32(false, a, false, b3,
                                                     (short)0, acc3, false, false);
    }

    // D layout: VGPR j holds row M = j + 8*hf, lane lr = column
    const size_t outBase = (size_t)rowTile * 16 * CRF_T;
    const float bz0 = bias[lr];
    const float bz1 = bias[16 + lr];
    const float bz2 = bias[32 + lr];
    const float bz3 = bias[48 + lr];
#pragma unroll
    for (int j = 0; j < 8; ++j) {
        const size_t r = outBase + (size_t)(j + 8 * hf) * CRF_T;
        em[r + lr]      = acc0[j] + bz0;
        em[r + 16 + lr] = acc1[j] + bz1;
        em[r + 32 + lr] = acc2[j] + bz2;
        em[r + 48 + lr] = acc3[j] + bz3;
    }
}

// -------------------- Kernel 2: numerator per batch ------------------------
__global__ void crf_numerator(const float* __restrict__ em,     // (B,S,T)
                              const float* __restrict__ trans,  // (T,T)
                              const int*   __restrict__ y,      // (B,S)
                              float* __restrict__ numer)        // (B)
{
    __shared__ float red[256];
    const int b   = blockIdx.x;
    const int tid = threadIdx.x;
    const int* __restrict__ yb = y + (size_t)b * CRF_S;
    const float* __restrict__ emB = em + (size_t)b * CRF_S * CRF_T;

    float acc = 0.0f;
    for (int s = tid; s < CRF_S; s += 256) {
        const int ys = yb[s];
        acc += emB[(size_t)s * CRF_T + ys];
        if (s > 0) acc += trans[yb[s - 1] * CRF_T + ys];
    }
    red[tid] = acc;
    __syncthreads();
    for (int off = 128; off > 0; off >>= 1) {
        if (tid < off) red[tid] += red[tid + off];
        __syncthreads();
    }
    if (tid == 0) numer[b] = red[0];
}

// -------------------- Kernel 3: forward scan + finalize --------------------
// One block per batch element. 256 threads: t = tid&63 (target tag),
// q = tid>>6 (quarter of the 64-wide p-reduction).
__global__ void crf_forward_scan(const float* __restrict__ em,     // (B,S,T)
                                 const float* __restrict__ trans,  // (T,T)
                                 const float* __restrict__ numer,  // (B)
                                 float* __restrict__ out)          // scalar
{
    __shared__ float tr[CRF_T * CRF_T];   // 16 KB
    __shared__ float lp[CRF_T];
    __shared__ float redm[4][CRF_T];
    __shared__ float reds[4][CRF_T];

    const int b   = blockIdx.x;
    const int tid = threadIdx.x;
    const int t   = tid & 63;
    const int q   = tid >> 6;

    for (int i = tid; i < CRF_T * CRF_T; i += 256) tr[i] = trans[i];

    const float* __restrict__ emB = em + (size_t)b * CRF_S * CRF_T;
    if (q == 0) lp[t] = emB[t];                       // s = 0
    __syncthreads();

    const int pbase = q * 16;
    for (int s = 1; s < CRF_S; ++s) {
        // pass 1: partial max over this quarter's p range
        float m = -__builtin_inff();
#pragma unroll
        for (int i = 0; i < 16; ++i) {
            const int p = pbase + i;
            m = fmaxf(m, lp[p] + tr[p * CRF_T + t]);
        }
        redm[q][t] = m;
        __syncthreads();
        const float mm = fmaxf(fmaxf(redm[0][t], redm[1][t]),
                               fmaxf(redm[2][t], redm[3][t]));
        // pass 2: partial sum of exp(v - mm)
        float ssum = 0.0f;
#pragma unroll
        for (int i = 0; i < 16; ++i) {
            const int p = pbase + i;
            ssum += __expf(lp[p] + tr[p * CRF_T + t] - mm);
        }
        reds[q][t] = ssum;
        __syncthreads();
        if (q == 0) {
            const float stot = reds[0][t] + reds[1][t] + reds[2][t] + reds[3][t];
            lp[t] = mm + __logf(stot) + emB[(size_t)s * CRF_T + t];
        }
        __syncthreads();
    }

    // denominator = LSE over final lp; fold into scalar output
    if (tid == 0) {
        float m = lp[0];
        for (int i = 1; i < CRF_T; ++i) m = fmaxf(m, lp[i]);
        float ssum = 0.0f;
        for (int i = 0; i < CRF_T; ++i) ssum += __expf(lp[i] - m);
        const float denom = m + __logf(ssum);
        atomicAdd(out, denom - numer[b]);
    }
}

// ---------------------------------------------------------------------------
extern "C" void kernel_launch(void* const* d_in, const int* in_sizes, int n_in,
                              void* d_out, int out_size, void* d_ws, size_t ws_size,
                              hipStream_t stream) {
    (void)in_sizes; (void)n_in; (void)out_size; (void)ws_size;

    const float* x     = (const float*)d_in[0];  // (B,S,NIN)
    const float* W     = (const float*)d_in[1];  // (NIN,T)
    const float* bias  = (const float*)d_in[2];  // (T)
    const float* trans = (const float*)d_in[3];  // (T,T)
    const int*   y     = (const int*)  d_in[4];  // (B,S) int32

    float* em    = (float*)d_ws;                          // 65536*64 floats (16 MB)
    float* numer = em + (size_t)CRF_ROWS * CRF_T;         // 128 floats

    // 1) emissions GEMM: 4096 row tiles, 8 waves/block
    crf_emissions_wmma<<<dim3(CRF_ROWS / 16 / 8), dim3(256), 0, stream>>>(
        x, W, bias, em);

    // 2) per-batch numerator
    crf_numerator<<<dim3(CRF_B), dim3(256), 0, stream>>>(em, trans, y, numer);

    // 3) zero scalar output, then forward scan + finalize
    hipMemsetAsync(d_out, 0, sizeof(float), stream);
    crf_forward_scan<<<dim3(CRF_B), dim3(256), 0, stream>>>(em, trans, numer,
                                                            (float*)d_out);
}